// TensorTrainLayer_45543833206790
// MI455X (gfx1250) — compile-verified
//
#include <hip/hip_runtime.h>

// ---- CDNA5 (gfx1250) wave32 WMMA types ----
typedef float v2f __attribute__((ext_vector_type(2)));
typedef float v8f __attribute__((ext_vector_type(8)));

#define BATCH 4096
#define NCAR  8
#define FDIM  64
#define DDIM  64
#define CDIM  10
#define SPB   16          // samples per workgroup (WMMA row dim)
#define LSTR  68          // padded inner stride (floats): 68%64==4 -> conflict-free b64 lanes
#define SROW  (16 * LSTR) // floats per sample-row group (16 rows of 68)

// D = A(16x4) * B(4x16) + C, fp32 in/out.  A: v2f per lane, B: v2f per lane, C/D: v8f.
__device__ __forceinline__ v8f wmma4(v2f a, v2f b, v8f c) {
    return __builtin_amdgcn_wmma_f32_16x16x4_f32(
        /*neg_a=*/false, a, /*neg_b=*/false, b,
        /*c_mod=*/(short)0, c, /*reuse_a=*/false, /*reuse_b=*/false);
}

__global__ __launch_bounds__(128)
void tt_forward_kernel(const float* __restrict__ x,          // (B, 8, 64)
                       const float* __restrict__ core_first, // (10, 64, 64)
                       const float* __restrict__ cores_mid,  // (6, 64, 64, 64)
                       const float* __restrict__ core_last,  // (64, 64)
                       float* __restrict__ out)              // (B, 10)
{
    // T double buffer: T[s][c(16)][r(64, pad 68)];  M: M[s][r(16)][l(64, pad 68)]
    __shared__ float ldsTa[SPB * SROW];
    __shared__ float ldsTb[SPB * SROW];
    __shared__ float ldsM [SPB * SROW];
    __shared__ float ldsX [SPB * LSTR];  // x[s][p(64, pad 68)]

    const int tid   = threadIdx.x;
    const int wave  = tid >> 5;
    const int lane  = tid & 31;
    const int half  = lane >> 4;   // lane group: selects K-pair / output row half
    const int lm    = lane & 15;
    const int sbase = blockIdx.x * SPB;

    // ---- stage x[:, car, :] for this block's 16 samples into LDS ----
    auto stage_x = [&](int car) {
        int s  = tid >> 3;           // 0..15
        int p0 = (tid & 7) * 8;      // 0..56
        const float4* g =
            reinterpret_cast<const float4*>(x + (size_t)(sbase + s) * (NCAR * FDIM) + car * FDIM + p0);
        float4 v0 = g[0];
        float4 v1 = g[1];
        float4* l = reinterpret_cast<float4*>(ldsX + s * LSTR + p0);
        l[0] = v0; l[1] = v1;
    };

    // ================= carriage 0: T0[s,c,r] = sum_p x0[s,p]*A1[c,p,r] =================
    stage_x(0);
    __syncthreads();
    for (int rb = 0; rb < 4; ++rb) {
        const int rbase = rb * 16;
        for (int j = 0; j < 4; ++j) {
            const int c = wave + 4 * j;              // wave-uniform tile (class row block)
            v8f acc = {0.f, 0.f, 0.f, 0.f, 0.f, 0.f, 0.f, 0.f};
            if (c < CDIM) {                          // uniform branch: EXEC stays full
                const float* bptr = core_first + (size_t)c * FDIM * DDIM + rbase + lm;
#pragma unroll
                for (int kk = 0; kk < 16; ++kk) {
                    const int p0 = kk * 4 + 2 * half;
                    v2f a = *reinterpret_cast<const v2f*>(ldsX + lm * LSTR + p0);
                    v2f b;
                    b.x = bptr[(size_t)p0 * DDIM];
                    b.y = bptr[(size_t)(p0 + 1) * DDIM];
                    acc = wmma4(a, b, acc);
                }
            }
#pragma unroll
            for (int v = 0; v < 8; ++v) {            // D-tile: row = v + 8*half (= sample)
                const int s = v + 8 * half;
                ldsTa[s * SROW + c * LSTR + rbase + lm] = acc[v];
            }
        }
    }
    __syncthreads();

    // ================= carriages 1..6: T <- T * M(x_i, core_i) =================
    float* Told = ldsTa;
    float* Tnew = ldsTb;
    for (int step = 1; step <= NCAR - 2; ++step) {
        stage_x(step);
        __syncthreads();
        const float* corep = cores_mid + (size_t)(step - 1) * (DDIM * FDIM * DDIM);

        for (int rb = 0; rb < 4; ++rb) {
            const int rbase = rb * 16;

            // ---- phase 1: M[s, l, r-block] = X @ core  (rows=samples, K=p) ----
            for (int j = 0; j < 16; ++j) {
                const int l = wave * 16 + j;                 // wave-uniform
                const float* bptr = corep + (size_t)l * FDIM * DDIM + rbase + lm;
                if (j + 1 < 16)
                    __builtin_prefetch(corep + (size_t)(l + 1) * FDIM * DDIM + rbase, 0, 1);
                v8f acc = {0.f, 0.f, 0.f, 0.f, 0.f, 0.f, 0.f, 0.f};
#pragma unroll
                for (int kk = 0; kk < 16; ++kk) {
                    const int p0 = kk * 4 + 2 * half;
                    v2f a = *reinterpret_cast<const v2f*>(ldsX + lm * LSTR + p0);
                    v2f b;
                    b.x = bptr[(size_t)p0 * DDIM];
                    b.y = bptr[(size_t)(p0 + 1) * DDIM];
                    acc = wmma4(a, b, acc);
                }
#pragma unroll
                for (int v = 0; v < 8; ++v) {                // scatter: M[s][r][l]
                    const int s = v + 8 * half;
                    ldsM[s * SROW + lm * LSTR + l] = acc[v];
                }
            }
            __syncthreads();

            // ---- phase 2: Tnew[s, c, r-block] = Told[s] @ M[s]  (per-sample, K=l) ----
            for (int q = 0; q < 4; ++q) {
                const int s = wave + 4 * q;                  // wave-uniform
                v8f acc = {0.f, 0.f, 0.f, 0.f, 0.f, 0.f, 0.f, 0.f};
#pragma unroll
                for (int kk = 0; kk < 16; ++kk) {
                    const int l0 = kk * 4 + 2 * half;
                    v2f a = *reinterpret_cast<const v2f*>(Told + s * SROW + lm * LSTR + l0);
                    v2f b = *reinterpret_cast<const v2f*>(ldsM + s * SROW + lm * LSTR + l0);
                    acc = wmma4(a, b, acc);
                }
#pragma unroll
                for (int v = 0; v < 8; ++v) {
                    const int c = v + 8 * half;
                    Tnew[s * SROW + c * LSTR + rbase + lm] = acc[v];
                }
            }
            __syncthreads();
        }
        float* t = Told; Told = Tnew; Tnew = t;
    }

    // ================= carriage 7: m[s,l] = sum_p x7[s,p]*AN[l,p]; out = <T, m> =================
    stage_x(NCAR - 1);
    __syncthreads();
    {
        const int lbase = wave * 16;                         // one l-tile per wave
        const float* bptr = core_last + (size_t)(lbase + lm) * FDIM;
        v8f acc = {0.f, 0.f, 0.f, 0.f, 0.f, 0.f, 0.f, 0.f};
#pragma unroll
        for (int kk = 0; kk < 16; ++kk) {
            const int p0 = kk * 4 + 2 * half;
            v2f a = *reinterpret_cast<const v2f*>(ldsX + lm * LSTR + p0);
            v2f b = *reinterpret_cast<const v2f*>(bptr + p0); // contiguous (l row of AN)
            acc = wmma4(a, b, acc);
        }
#pragma unroll
        for (int v = 0; v < 8; ++v) {                        // m[s][l] compact in ldsM
            const int s = v + 8 * half;
            ldsM[s * LSTR + lbase + lm] = acc[v];
        }
    }
    __syncthreads();

    for (int o = tid; o < SPB * CDIM; o += 128) {
        const int s = o / CDIM;
        const int c = o % CDIM;
        const float* trow = Told + s * SROW + c * LSTR;
        const float* mrow = ldsM + s * LSTR;
        float sum = 0.f;
#pragma unroll
        for (int l = 0; l < DDIM; ++l) sum += trow[l] * mrow[l];
        out[(size_t)(sbase + s) * CDIM + c] = sum;
    }
}

extern "C" void kernel_launch(void* const* d_in, const int* in_sizes, int n_in,
                              void* d_out, int out_size, void* d_ws, size_t ws_size,
                              hipStream_t stream) {
    const float* x          = (const float*)d_in[0];
    const float* core_first = (const float*)d_in[1];
    const float* cores_mid  = (const float*)d_in[2];
    const float* core_last  = (const float*)d_in[3];
    float* out = (float*)d_out;
    (void)in_sizes; (void)n_in; (void)out_size; (void)d_ws; (void)ws_size;

    dim3 grid(BATCH / SPB);   // 256 workgroups
    dim3 block(128);          // 4 wave32 waves
    hipLaunchKernelGGL(tt_forward_kernel, grid, block, 0, stream,
                       x, core_first, cores_mid, core_last, out);
}